// Fusion_69887707840663
// MI455X (gfx1250) — compile-verified
//
#include <hip/hip_runtime.h>
#include <math.h>

// Problem geometry (fixed by the reference)
#define NVIEW 8
#define NC    32
#define IH    128
#define IW    128
#define ND    64
#define NH    64
#define NW    64

// LDS layout (float words).
#define R_XSTRIDE 33
#define R_VSTRIDE (IW * R_XSTRIDE)                 // 4224 words / view
#define STAGE_OFF (NVIEW * R_VSTRIDE)              // 33792 words (132 KB)
#define STAGE_BUF_WORDS (2 * NC * IW)              // 8192 words (32 KB) per buffer
#define TILE_OFF  STAGE_OFF                        // stage & tiles are disjoint in time
#define TILE_WSTRIDE (32 * R_XSTRIDE)              // 1056 words / wave
#define SMEM_WORDS (STAGE_OFF + 16 * TILE_WSTRIDE) // 50688 words = 202752 B (<320KB)
// 2 stage buffers = 16384 words <= 16*1056 = 16896 words of tile space. OK.

#if __has_builtin(__builtin_amdgcn_tensor_load_to_lds)
#define USE_TDM 1
#else
#define USE_TDM 0
#endif

typedef unsigned int v4u   __attribute__((ext_vector_type(4)));
typedef int          v8i_t __attribute__((ext_vector_type(8)));
typedef int          v4i_t __attribute__((ext_vector_type(4)));

#if USE_TDM
// Issue one TDM load: tile = [x:128][y:2][c:2] f32 from global into LDS.
// tensor_dim1 = rows remaining below yr0 so the y==1 row reads zeros when
// yr0 == 127 (matches the zero-padded corner whose weight is already 0).
__device__ __forceinline__ void tdm_issue(const float* gbase, unsigned lds_byte,
                                          unsigned tdim1) {
  unsigned long long ga = (unsigned long long)(const void*)gbase;
  v4u g0;
  g0.x = 1u;                                   // count=1 (valid user descriptor)
  g0.y = lds_byte;                             // lds_addr (bytes)
  g0.z = (unsigned)ga;                         // global_addr[31:0]
  g0.w = ((unsigned)(ga >> 32) & 0x01FFFFFFu)  // global_addr[56:32]
         | (2u << 30);                         // type=2 ("image")
  v8i_t g1;
  g1[0] = (int)(2u << 16);                     // data_size=2 (4 bytes)
  g1[1] = (int)(128u << 16);                   // tensor_dim0[15:0]=128
  g1[2] = (int)(tdim1 << 16);                  // tensor_dim0[31:16]=0 | tensor_dim1[15:0]
  g1[3] = (int)(128u << 16);                   // tensor_dim1 hi=0 | tile_dim0=128
  g1[4] = (int)(2u | (2u << 16));              // tile_dim1=2 (y), tile_dim2=2 (c)
  g1[5] = 128;                                 // tensor_dim0_stride = 128 elems (y step)
  g1[6] = (int)((unsigned)(IH * IW) << 16);    // t0s hi=0 | tensor_dim1_stride[15:0]=16384
  g1[7] = 0;                                   // tensor_dim1_stride[47:16] = 0
  v4i_t g2;
  g2[0] = NC;                                  // tensor_dim2 (never exceeded by tile z)
  g2[1] = 0; g2[2] = 0; g2[3] = 0;             // tensor_dim3 / stride2 / tile_dim3 unused
  v4i_t g3;
  g3[0] = 0; g3[1] = 0; g3[2] = 0; g3[3] = 0;  // dim3 stride / dim4 unused
  v8i_t g4;                                    // 6-arg toolchain extension operand:
  g4[0] = 0; g4[1] = 0; g4[2] = 0; g4[3] = 0;  // zero-filled (inert / NULL descriptor
  g4[4] = 0; g4[5] = 0; g4[6] = 0; g4[7] = 0;  // extension, probe-verified form)
  __builtin_amdgcn_tensor_load_to_lds(g0, g1, g2, g3, g4, 0);
}

__device__ __forceinline__ void wait_tensor_le1() {
#if __has_builtin(__builtin_amdgcn_s_wait_tensorcnt)
  __builtin_amdgcn_s_wait_tensorcnt(1);
#else
  asm volatile("s_wait_tensorcnt 0x1" ::: "memory");
#endif
}
__device__ __forceinline__ void wait_tensor_0() {
#if __has_builtin(__builtin_amdgcn_s_wait_tensorcnt)
  __builtin_amdgcn_s_wait_tensorcnt(0);
#else
  asm volatile("s_wait_tensorcnt 0x0" ::: "memory");
#endif
}
#else  // !USE_TDM fallback: per-lane async copy (ASYNCcnt-tracked)
__device__ __forceinline__ void async_load16_to_lds(unsigned lds_byte_addr,
                                                    const float* gaddr) {
  asm volatile("global_load_async_to_lds_b128 %0, %1, off"
               :: "v"(lds_byte_addr), "v"(gaddr)
               : "memory");
}
__device__ __forceinline__ void wait_async0() {
#if __has_builtin(__builtin_amdgcn_s_wait_asynccnt)
  __builtin_amdgcn_s_wait_asynccnt(0);
#else
  asm volatile("s_wait_asynccnt 0" ::: "memory");
#endif
}
#endif

__global__ __launch_bounds__(512)
void backproject_kernel(const float* __restrict__ xrays,   // [8,32,128,128]
                        const float* __restrict__ angles,  // [8]
                        float* __restrict__ out) {         // [32,64,64,64]
  extern __shared__ float smem[];
  const int tid   = threadIdx.x;
  const int wave  = tid >> 5;
  const int lane  = tid & 31;
  const int d     = blockIdx.x >> 2;   // d-slice
  const int chunk = blockIdx.x & 3;    // quarter of the 64x64 hw-plane

  // Per-view rotation (mirrors reference math exactly).
  float cs[NVIEW], sn[NVIEW];
#pragma unroll
  for (int v = 0; v < NVIEW; ++v) {
    float a = angles[v];
    cs[v] = cosf(a);
    sn[v] = sinf(a);
  }

  // ---- y (detector row) geometry: depends only on d (wave-uniform) ----
  const float zd  = -120.0f + (float)d * (240.0f / 63.0f);
  const float gy  = 2.0f * zd / 240.0f;
  const float ys  = (gy + 1.0f) * 0.5f * 127.0f;
  const float y0f = floorf(ys);
  const float fy1 = ys - y0f;
  const float fy0 = 1.0f - fy1;
  const int   iy0 = (int)y0f;
  const int   iy1 = iy0 + 1;
  const float my0 = (iy0 >= 0 && iy0 <= IH - 1) ? fy0 : 0.0f;  // per-corner zero pad
  const float my1 = (iy1 >= 0 && iy1 <= IH - 1) ? fy1 : 0.0f;
  const int   yr0 = iy0 < 0 ? 0 : (iy0 > IH - 1 ? IH - 1 : iy0);
#if !USE_TDM
  const int   yr1 = iy1 < 0 ? 0 : (iy1 > IH - 1 ? IH - 1 : iy1);
#endif

  // ---- Phase A: build R[v][x][c] = my0*img[v,c,y0,x] + my1*img[v,c,y1,x] ----
  // Stage layout per buffer: [c:32][j:2][x:128] contiguous (TDM z,y,x loop order).
#if USE_TDM
  const unsigned tdim1 = (unsigned)(IH - yr0);
  // Each wave DMAs its own 2 channels: tile 128 x 2 x 2 = 2KB, all 16 waves
  // cover c=0..31. Double-buffered: issue view v+1 while combining view v.
  {
    const float* g0 = xrays + (((0 * NC + wave * 2) * IH + yr0) * IW);
    unsigned l0 = (unsigned)(unsigned long long)
        (smem + STAGE_OFF + (wave * 2) * (2 * IW));
    tdm_issue(g0, l0, tdim1);
  }
  for (int v = 0; v < NVIEW; ++v) {
    if (v + 1 < NVIEW) {
      const float* gn = xrays + ((((v + 1) * NC + wave * 2) * IH + yr0) * IW);
      unsigned ln = (unsigned)(unsigned long long)
          (smem + STAGE_OFF + ((v + 1) & 1) * STAGE_BUF_WORDS + (wave * 2) * (2 * IW));
      tdm_issue(gn, ln, tdim1);
      wait_tensor_le1();   // in-order per wave: view v's tile is resident
    } else {
      wait_tensor_0();
    }
    __syncthreads();       // all waves' tiles for view v are in LDS

    const int sb = STAGE_OFF + (v & 1) * STAGE_BUF_WORDS;
#pragma unroll
    for (int it = 0; it < 8; ++it) {
      int idx = it * 512 + tid;      // 0..4095
      int c  = idx >> 7;
      int xp = idx & 127;
      float r0 = smem[sb + (c * 2 + 0) * IW + xp];
      float r1 = smem[sb + (c * 2 + 1) * IW + xp];
      smem[v * R_VSTRIDE + xp * R_XSTRIDE + c] = my0 * r0 + my1 * r1;
    }
    __syncthreads();       // buffer v&1 may now be overwritten next iter
  }
#else
  for (int v = 0; v < NVIEW; ++v) {
#pragma unroll
    for (int it = 0; it < 4; ++it) {
      int idx = it * 512 + tid;      // 0..2047 16B chunks
      int l16 = idx & 31;
      int row = idx >> 5;            // 0..63
      int c   = row >> 1;
      int j   = row & 1;
      int yy  = j ? yr1 : yr0;
      const float* g = xrays + (((v * NC + c) * IH + yy) * IW) + (l16 << 2);
      unsigned laddr = (unsigned)(unsigned long long)
          (smem + STAGE_OFF + ((c * 2 + j) * IW) + (l16 << 2));
      async_load16_to_lds(laddr, g);
    }
    wait_async0();
    __syncthreads();
#pragma unroll
    for (int it = 0; it < 8; ++it) {
      int idx = it * 512 + tid;
      int c  = idx >> 7;
      int xp = idx & 127;
      float r0 = smem[STAGE_OFF + (c * 2 + 0) * IW + xp];
      float r1 = smem[STAGE_OFF + (c * 2 + 1) * IW + xp];
      smem[v * R_VSTRIDE + xp * R_XSTRIDE + c] = my0 * r0 + my1 * r1;
    }
    __syncthreads();
  }
#endif

  // ---- Phase B: sweep (h,w); lane = channel; x-index math is wave-uniform ----
  for (int t = 0; t < 2; ++t) {
    const int tileIdx = chunk * 32 + wave * 2 + t;  // 0..127
    const int h  = tileIdx >> 1;
    const int w0 = (tileIdx & 1) * 32;
    const float yh = -120.0f + (float)h * (240.0f / 63.0f);

    for (int p = 0; p < 32; ++p) {
      const int w = w0 + p;
      const float xw = -120.0f + (float)w * (240.0f / 63.0f);
      float acc = 0.0f;
#pragma unroll
      for (int v = 0; v < NVIEW; ++v) {
        float gx  = 2.0f * (cs[v] * xw - sn[v] * yh) / 240.0f;
        float xs  = (gx + 1.0f) * 0.5f * 127.0f;
        float x0f = floorf(xs);
        float fx1 = xs - x0f;
        float fx0 = 1.0f - fx1;
        int ix0 = (int)x0f;
        int ix1 = ix0 + 1;
        float m0 = (ix0 >= 0 && ix0 <= IW - 1) ? fx0 : 0.0f;
        float m1 = (ix1 >= 0 && ix1 <= IW - 1) ? fx1 : 0.0f;
        int xc0 = ix0 < 0 ? 0 : (ix0 > IW - 1 ? IW - 1 : ix0);
        int xc1 = ix1 < 0 ? 0 : (ix1 > IW - 1 ? IW - 1 : ix1);
        const float* Rv = smem + v * R_VSTRIDE;
        acc = fmaf(m0, Rv[xc0 * R_XSTRIDE + lane], acc);  // uniform addr + lane: no conflicts
        acc = fmaf(m1, Rv[xc1 * R_XSTRIDE + lane], acc);
      }
      smem[TILE_OFF + wave * TILE_WSTRIDE + lane * R_XSTRIDE + p] = acc * 0.125f;
    }

    // Wave-local transpose -> fully coalesced 128B stores along w.
#pragma unroll 4
    for (int c2 = 0; c2 < NC; ++c2) {
      float val = smem[TILE_OFF + wave * TILE_WSTRIDE + c2 * R_XSTRIDE + lane];
      out[(((c2 * ND + d) * NH + h) * NW) + w0 + lane] = val;
    }
  }
}

extern "C" void kernel_launch(void* const* d_in, const int* in_sizes, int n_in,
                              void* d_out, int out_size, void* d_ws, size_t ws_size,
                              hipStream_t stream) {
  const float* xrays  = (const float*)d_in[0];  // [1,8,32,128,128] f32
  const float* angles = (const float*)d_in[1];  // [1,8] f32
  float* out = (float*)d_out;                   // [1,32,64,64,64] f32

  const size_t shmem = (size_t)SMEM_WORDS * sizeof(float);  // 202752 B
  (void)hipFuncSetAttribute(reinterpret_cast<const void*>(backproject_kernel),
                            hipFuncAttributeMaxDynamicSharedMemorySize, (int)shmem);

  backproject_kernel<<<dim3(ND * 4), dim3(512), shmem, stream>>>(xrays, angles, out);
}